// MatrixTree_68324339745149
// MI455X (gfx1250) — compile-verified
//
#include <hip/hip_runtime.h>
#include <math.h>

// Problem sizes (fixed by the reference)
#define BATCH 64
#define NN    512
#define BK    16
#define NSTEP (NN / BK)   // 32 block steps
#define TPB   256         // 8 wave32 waves per workgroup

typedef __attribute__((ext_vector_type(2))) float v2f;
typedef __attribute__((ext_vector_type(8))) float v8f;

#if defined(__has_builtin)
#  if __has_builtin(__builtin_amdgcn_wmma_f32_16x16x4_f32)
#    define HAVE_WMMA_F32X4 1
#  endif
#endif

// ---------------------------------------------------------------------------
// Kernel 1: column sums of masked exp(x):  cs[b][j] = sum_{i<len, i!=j} e^{x[b][i][j]}
// Lanes map to consecutive j -> every global read of x[b][i][j] is coalesced.
// ---------------------------------------------------------------------------
__global__ void colsum_kernel(const float* __restrict__ x,
                              const int* __restrict__ lengths,
                              float* __restrict__ cs) {
    int id = blockIdx.x * blockDim.x + threadIdx.x;   // over B*N
    int b = id >> 9;
    int j = id & (NN - 1);
    int len = lengths[b];
    const float* xb = x + (size_t)b * NN * NN;
    float s = 0.f;
    for (int i = 0; i < NN; ++i) {
        float e = __expf(xb[i * NN + j]);
        s += (i < len && j < len && i != j) ? e : 0.f;
    }
    cs[id] = s;
}

// ---------------------------------------------------------------------------
// Kernel 2: materialize the Laplacian L per the reference:
//   row 0            : d[j] = mask_j ? exp(x[j][j]) : 0
//   diag (j>0)       : mask_j ? colsum[j] : 1.0      (padded diag = identity)
//   off-diag         : (mask_i && mask_j) ? -exp(x[i][j]) : 0
// ---------------------------------------------------------------------------
__global__ void buildL_kernel(const float* __restrict__ x,
                              const int* __restrict__ lengths,
                              const float* __restrict__ cs,
                              float* __restrict__ L) {
    size_t id = (size_t)blockIdx.x * blockDim.x + threadIdx.x;  // over B*N*N
    int b   = (int)(id >> 18);
    int rem = (int)(id & (NN * NN - 1));
    int i = rem >> 9;
    int j = rem & (NN - 1);
    int len = lengths[b];
    const float* xb = x + (size_t)b * NN * NN;
    float v;
    if (i == 0) {
        v = (j < len) ? __expf(xb[j * NN + j]) : 0.f;
    } else if (i == j) {
        v = (j < len) ? cs[b * NN + j] : 1.f;
    } else {
        v = (i < len && j < len) ? -__expf(xb[i * NN + j]) : 0.f;
    }
    L[id] = v;
}

// ---------------------------------------------------------------------------
// Kernel 3: in-place blocked Gauss-Jordan inversion (no pivoting), one
// workgroup per batch matrix.  BK=16 block steps; trailing rank-16 update
// runs on V_WMMA_F32_16X16X4_F32 (4 chained WMMAs per 16x16 tile).
//
// LDS (exactly 64KB):
//   Ct[k][i] = L[i][kb+k]   column panel, transposed (conflict-free A reads;
//                           pivot block overlays Ct[*][kb..kb+15])
//   Rp[k][j] = -Rnew[k][j]  row panel; stored NEGATED so the trailing WMMA
//                           is D = C + A*(-R) with no operand negation
// ---------------------------------------------------------------------------
__global__ void __launch_bounds__(TPB)
invert_kernel(float* __restrict__ Lg) {
    __shared__ float Ct[BK][NN];   // 32 KB
    __shared__ float Rp[BK][NN];   // 32 KB

    float* L = Lg + (size_t)blockIdx.x * NN * NN;
    const int tid  = threadIdx.x;
    const int lane = tid & 31;
    const int wid  = tid >> 5;       // 0..7
    const int half = lane >> 4;      // 0 or 1
    const int ln   = lane & 15;      // 0..15

    for (int st = 0; st < NSTEP; ++st) {
        const int kb = st * BK;

        // ---- load panels (b128 global reads) -------------------------------
        for (int idx = tid; idx < NN * 4; idx += TPB) {
            // column panel: one float4 (4 consecutive k) of one row per thread
            int i = idx >> 2, q = idx & 3;
            float4 v = *(const float4*)&L[i * NN + kb + 4 * q];
            Ct[4 * q + 0][i] = v.x;
            Ct[4 * q + 1][i] = v.y;
            Ct[4 * q + 2][i] = v.z;
            Ct[4 * q + 3][i] = v.w;
        }
        for (int idx = tid; idx < (BK * NN) / 4; idx += TPB) {
            int k = idx >> 7, j4 = idx & 127;
            *(float4*)&Rp[k][4 * j4] = *(const float4*)&L[(kb + k) * NN + 4 * j4];
        }
        __syncthreads();

        // ---- invert 16x16 pivot block in place (stored as P^T in Ct) -------
        // After this, Ct[r][kb+c] = (P^T)^{-1}[r][c]  =>  Pinv[k][t] = Ct[t][kb+k]
        {
            const int r = tid >> 4, c = tid & 15;   // 256 threads, 1 elem each
            for (int k = 0; k < BK; ++k) {
                float p    = Ct[k][kb + k];
                float ip   = 1.0f / p;
                float m_rc = Ct[r][kb + c];
                float m_rk = Ct[r][kb + k];
                float m_kc = Ct[k][kb + c];
                float nv;
                if (r == k)      nv = (c == k) ? ip : m_kc * ip;
                else if (c == k) nv = -m_rk * ip;
                else             nv = m_rc - m_rk * ip * m_kc;
                __syncthreads();                  // all reads done
                Ct[r][kb + c] = nv;
                __syncthreads();                  // writes visible
            }
        }

        // ---- Rnew = Pinv * Rold  (each column owned by one thread) ---------
        // LDS gets -Rnew; global row block gets +Rnew (Pinv inside block cols)
        for (int j = tid; j < NN; j += TPB) {
            float rold[BK];
#pragma unroll
            for (int t = 0; t < BK; ++t) rold[t] = Rp[t][j];
            float rnew[BK];
#pragma unroll
            for (int k = 0; k < BK; ++k) {
                float acc = 0.f;
#pragma unroll
                for (int t = 0; t < BK; ++t) acc += Ct[t][kb + k] * rold[t];
                rnew[k] = acc;
            }
            const bool inblk = (j >= kb) && (j < kb + BK);
#pragma unroll
            for (int k = 0; k < BK; ++k) {
                Rp[k][j] = -rnew[k];
                L[(kb + k) * NN + j] = inblk ? Ct[j - kb][kb + k] : rnew[k];
            }
        }
        __syncthreads();   // Rp now holds -Rnew for the WMMA trailing update

        // ---- Cnew = -Cold * Pinv  -> global column block (rows outside) ----
        for (int i = tid; i < NN; i += TPB) {
            if (i >= kb && i < kb + BK) continue;
#pragma unroll
            for (int k = 0; k < BK; ++k) {
                float acc = 0.f;
#pragma unroll
                for (int t = 0; t < BK; ++t) acc += Ct[t][i] * Ct[k][kb + t];
                L[i * NN + kb + k] = -acc;
            }
        }

        // ---- trailing rank-16 update: L += Cold * (-Rnew)  (WMMA f32) ------
        // Each wave owns whole tile rows: A operand loaded once per row.
        // All control flow is wave-uniform -> EXEC all-ones for WMMA.
        for (int ti = wid; ti < 32; ti += 8) {
            if (ti == st) continue;               // row panel handled above
            // A[m][k] = Cold[ti*16+m][k] = Ct[k][ti*16+m]; per-lane m = ln
            float a0[4], a1[4];
#pragma unroll
            for (int q = 0; q < 4; ++q) {
                a0[q] = Ct[4 * q + 2 * half + 0][ti * BK + ln];
                a1[q] = Ct[4 * q + 2 * half + 1][ti * BK + ln];
            }
            for (int tj = 0; tj < 32; ++tj) {
                if (tj == st) continue;           // column panel handled above
                float* Lt = L + (ti * BK) * NN + tj * BK;
                v8f c;
#pragma unroll
                for (int r = 0; r < 8; ++r) c[r] = Lt[(r + 8 * half) * NN + ln];
#if defined(HAVE_WMMA_F32X4)
#pragma unroll
                for (int q = 0; q < 4; ++q) {
                    v2f a, b;
                    a.x = a0[q];
                    a.y = a1[q];
                    // B[k][n] = -Rnew[k][tj*16+n]
                    b.x = Rp[4 * q + 2 * half + 0][tj * BK + ln];
                    b.y = Rp[4 * q + 2 * half + 1][tj * BK + ln];
                    c = __builtin_amdgcn_wmma_f32_16x16x4_f32(
                            /*neg_a=*/false, a, /*neg_b=*/false, b,
                            /*c_mod=*/(short)0, c, /*reuse_a=*/false, /*reuse_b=*/false);
                }
#else
                // layout-identical scalar fallback (m = r + 8*half, n = ln);
                // Rp is already negated, so accumulate with +=
#pragma unroll
                for (int r = 0; r < 8; ++r) {
                    float acc = c[r];
#pragma unroll
                    for (int k = 0; k < BK; ++k)
                        acc += Ct[k][ti * BK + r + 8 * half] * Rp[k][tj * BK + ln];
                    c[r] = acc;
                }
#endif
#pragma unroll
                for (int r = 0; r < 8; ++r) Lt[(r + 8 * half) * NN + ln] = c[r];
            }
        }

        __threadfence_block();   // make trailing stores visible workgroup-wide
        __syncthreads();         // before next step's panel loads
    }
}

// ---------------------------------------------------------------------------
// Kernel 4: assemble the output from exp(x) and inv = L^{-1}:
//   out[i][j] = e*inv[j][j]*(j!=0) - e*inv[j][i]*(i!=0) + (i==j)*e*inv[j][0]
//   zero outside the valid (len x len) block.
// ---------------------------------------------------------------------------
__global__ void output_kernel(const float* __restrict__ x,
                              const int* __restrict__ lengths,
                              const float* __restrict__ inv,
                              float* __restrict__ out) {
    size_t id = (size_t)blockIdx.x * blockDim.x + threadIdx.x;
    int b   = (int)(id >> 18);
    int rem = (int)(id & (NN * NN - 1));
    int i = rem >> 9;
    int j = rem & (NN - 1);
    int len = lengths[b];
    if (i >= len || j >= len) { out[id] = 0.f; return; }
    const float* invb = inv + (size_t)b * NN * NN;
    float e  = __expf(x[id]);
    float t1 = (j == 0) ? 0.f : e * invb[j * NN + j];   // inv[j][j]
    float t2 = (i == 0) ? 0.f : e * invb[j * NN + i];   // inv^T[i][j] = inv[j][i]
    float o  = t1 - t2;
    if (i == j) o += e * invb[j * NN];                  // roots: d[j]*inv[j][0]
    out[id] = o;
}

// ---------------------------------------------------------------------------
extern "C" void kernel_launch(void* const* d_in, const int* in_sizes, int n_in,
                              void* d_out, int out_size, void* d_ws, size_t ws_size,
                              hipStream_t stream) {
    (void)in_sizes; (void)n_in; (void)out_size; (void)ws_size;
    const float* x       = (const float*)d_in[0];
    const int*   lengths = (const int*)d_in[1];
    float*       out     = (float*)d_out;

    // workspace layout: [colsum: B*N floats][L / inv: B*N*N floats]
    float* cs = (float*)d_ws;
    float* L  = cs + (size_t)BATCH * NN;

    colsum_kernel<<<BATCH * NN / TPB, TPB, 0, stream>>>(x, lengths, cs);
    buildL_kernel<<<(BATCH * NN * NN) / TPB, TPB, 0, stream>>>(x, lengths, cs, L);
    invert_kernel<<<BATCH, TPB, 0, stream>>>(L);
    output_kernel<<<(BATCH * NN * NN) / TPB, TPB, 0, stream>>>(x, lengths, L, out);
}